// MLP_41609643164200
// MI455X (gfx1250) — compile-verified
//
#include <hip/hip_runtime.h>
#include <hip/hip_bf16.h>

typedef float v2f __attribute__((ext_vector_type(2)));
typedef float v8f __attribute__((ext_vector_type(8)));

#if defined(__has_builtin)
#if __has_builtin(__builtin_amdgcn_tanhf)
#define FAST_TANH(x) __builtin_amdgcn_tanhf(x)
#endif
#endif
#ifndef FAST_TANH
#define FAST_TANH(x) tanhf(x)
#endif

__device__ __forceinline__ float lane_gather(float v, int srcLane) {
  // wave32: ds_bpermute uses byte index, bits [6:2]
  return __int_as_float(__builtin_amdgcn_ds_bpermute(srcLane << 2, __float_as_int(v)));
}

#define TPW 8  // 16-point tiles per wave

__global__ void __launch_bounds__(256)
pinn_mlp16_wmma(const float* __restrict__ x,  const float* __restrict__ x2,
                const float* __restrict__ W1, const float* __restrict__ W2,
                const float* __restrict__ B1,
                const float* __restrict__ W3, const float* __restrict__ B3,
                const float* __restrict__ W4, const float* __restrict__ B4,
                const float* __restrict__ W5, const float* __restrict__ B5,
                const float* __restrict__ W6, const float* __restrict__ B6,
                const float* __restrict__ W7, const float* __restrict__ B7,
                const float* __restrict__ W8, const float* __restrict__ B8,
                const float* __restrict__ W9, const float* __restrict__ B9,
                const float* __restrict__ alphap,
                float* __restrict__ out, int npts)
{
  const int lane = threadIdx.x & 31;
  const int p    = lane & 15;        // point index within tile / output feature for A loads
  const int hi   = lane >> 4;        // lane half
  const int wave = blockIdx.x * (blockDim.x >> 5) + (threadIdx.x >> 5);

  const float alpha = alphap[0];
  const float b9    = B9[0];

  const float* Ws[6] = {W3, W4, W5, W6, W7, W8};
  const float* Bs[6] = {B3, B4, B5, B6, B7, B8};

  // Preload W^T in the WMMA A-operand layout (16x4 f32 chunks):
  // chunk k, VGPR j, lane: element A[M = p][K = 4k + j + 2*hi] = W[K][p]
  v2f   aW[6][4];
  float ab[6][8];   // alpha * bias, per C-layout VGPR row (feature = v + 8*hi)
  #pragma unroll
  for (int l = 0; l < 6; ++l) {
    #pragma unroll
    for (int k = 0; k < 4; ++k)
      #pragma unroll
      for (int j = 0; j < 2; ++j)
        aW[l][k][j] = Ws[l][(4 * k + j + 2 * hi) * 16 + p];
    #pragma unroll
    for (int v = 0; v < 8; ++v)
      ab[l][v] = alpha * Bs[l][v + 8 * hi];
  }

  float aw1[8], aw2[8], ab1[8], w9v[8];
  #pragma unroll
  for (int v = 0; v < 8; ++v) {
    const int f = v + 8 * hi;
    aw1[v] = alpha * W1[f];
    aw2[v] = alpha * W2[f];
    ab1[v] = alpha * B1[f];
    w9v[v] = W9[f];
  }

  const int ntiles = (npts + 15) >> 4;
  int tile = wave * TPW;

  #pragma unroll 1
  for (int t = 0; t < TPW; ++t, ++tile) {
    if (tile >= ntiles) break;
    const int  base = tile << 4;
    const bool ok   = (base + p) < npts;
    const float xp  = ok ? x [base + p] : 0.0f;
    const float x2p = ok ? x2[base + p] : 0.0f;

    // Layer 1 (rank-1): Act[f][p] = tanh(alpha*(x[p]W1[f] + x2[p]W2[f] + B1[f]))
    v8f act;
    #pragma unroll
    for (int v = 0; v < 8; ++v)
      act[v] = FAST_TANH(fmaf(xp, aw1[v], fmaf(x2p, aw2[v], ab1[v])));

    // 6 hidden layers: D = W^T x Act  (fp32 WMMA, 4 chained K=4 chunks)
    #pragma unroll
    for (int l = 0; l < 6; ++l) {
      v8f d = {};
      #pragma unroll
      for (int k = 0; k < 4; ++k) {
        // Build B chunk (4x16): VGPR j holds row 4k+j (lanes 0-15), row 4k+j+2 (lanes 16-31).
        v2f bk;
        #pragma unroll
        for (int j = 0; j < 2; ++j) {
          const int r1 = 4 * k + j;
          const int r2 = r1 + 2;
          float bv;
          if (r2 < 8) {
            // both rows live in lanes 0-15 of Act; copy r2 into the upper half
            float g = lane_gather(act[r2], p);
            bv = hi ? g : act[r1];
          } else {
            // both rows live in lanes 16-31 of Act; copy r1 into the lower half
            float g = lane_gather(act[r1 & 7], p + 16);
            bv = hi ? act[r2 & 7] : g;
          }
          bk[j] = bv;
        }
        d = __builtin_amdgcn_wmma_f32_16x16x4_f32(false, aW[l][k], false, bk,
                                                  (short)0, d, false, false);
      }
      #pragma unroll
      for (int v = 0; v < 8; ++v)
        act[v] = FAST_TANH(fmaf(alpha, d[v], ab[l][v]));
    }

    // Output layer: out[p] = sum_f Act[f][p]*W9[f] + B9
    float partial = 0.0f;
    #pragma unroll
    for (int v = 0; v < 8; ++v)
      partial = fmaf(act[v], w9v[v], partial);
    const float other = lane_gather(partial, lane ^ 16);
    const float res   = partial + other + b9;
    if (hi == 0 && ok)
      out[base + p] = res;
  }
}

extern "C" void kernel_launch(void* const* d_in, const int* in_sizes, int n_in,
                              void* d_out, int out_size, void* d_ws, size_t ws_size,
                              hipStream_t stream) {
  (void)n_in; (void)d_ws; (void)ws_size; (void)out_size;
  const float* x   = (const float*)d_in[0];
  const float* x2  = (const float*)d_in[1];
  const float* W1  = (const float*)d_in[2];
  const float* W2  = (const float*)d_in[3];
  const float* B1  = (const float*)d_in[4];
  const float* W3  = (const float*)d_in[5];
  const float* B3  = (const float*)d_in[6];
  const float* W4  = (const float*)d_in[7];
  const float* B4  = (const float*)d_in[8];
  const float* W5  = (const float*)d_in[9];
  const float* B5  = (const float*)d_in[10];
  const float* W6  = (const float*)d_in[11];
  const float* B6  = (const float*)d_in[12];
  const float* W7  = (const float*)d_in[13];
  const float* B7  = (const float*)d_in[14];
  const float* W8  = (const float*)d_in[15];
  const float* B8  = (const float*)d_in[16];
  const float* W9  = (const float*)d_in[17];
  const float* B9  = (const float*)d_in[18];
  const float* alp = (const float*)d_in[19];
  float* out = (float*)d_out;

  const int npts   = in_sizes[0];
  const int ntiles = (npts + 15) >> 4;
  const int waves  = (ntiles + TPW - 1) / TPW;
  const int blocks = (waves + 7) / 8;   // 8 waves (256 threads) per block

  pinn_mlp16_wmma<<<blocks, 256, 0, stream>>>(
      x, x2, W1, W2, B1, W3, B3, W4, B4, W5, B5, W6, B6,
      W7, B7, W8, B8, W9, B9, alp, out, npts);
}